// TextGuidedAttention_64433099375101
// MI455X (gfx1250) — compile-verified
//
#include <hip/hip_runtime.h>
#include <math.h>

#define B_   16
#define C_   256
#define T_   4096
#define DT_  512
#define NH_  8
#define HD_  32
#define EPS_ 1e-5f

typedef __attribute__((ext_vector_type(2))) float v2f;
typedef __attribute__((ext_vector_type(8))) float v8f;

// ---------------------------------------------------------------------------
// K1: text path (per-batch block).  t1 = gelu(F_text@tp_w1+b1); t = t1@tp_w2+b2;
// q = t@q_w+q_b; w_qk[h][c] = sum_j q[h,j]*k_w[c, h*32+j]; b_qk[h] = q[h]·k_b[h-slice];
// gate: a = softmax2(dot(F_u,fa_w), dot(F_t,fa_w)); g = sigmoid((a0*F_u+a1*F_t)@gp_w + gp_b)
// ---------------------------------------------------------------------------
__global__ __launch_bounds__(256) void k1_text(
    const float* __restrict__ F_text, const float* __restrict__ F_unified,
    const float* __restrict__ F_teacher,
    const float* __restrict__ tp_w1, const float* __restrict__ tp_b1,
    const float* __restrict__ tp_w2, const float* __restrict__ tp_b2,
    const float* __restrict__ q_w,  const float* __restrict__ q_b,
    const float* __restrict__ k_w,  const float* __restrict__ k_b,
    const float* __restrict__ fa_w, const float* __restrict__ fa_b,
    const float* __restrict__ gp_w, const float* __restrict__ gp_b,
    float* __restrict__ ws_wqk, float* __restrict__ ws_bqk, float* __restrict__ ws_g)
{
    const int b   = blockIdx.x;
    const int tid = threadIdx.x;          // 256 threads; tid == output channel c
    __shared__ float s_t1[C_];
    __shared__ float s_t[C_];
    __shared__ float s_q[C_];
    __shared__ float s_fs[2];

    // --- t1 = gelu(F_text @ tp_w1 + b1), exact gelu ---
    {
        const float* ft = F_text + b * DT_;
        float acc = tp_b1[tid];
        for (int d = 0; d < DT_; ++d) acc += ft[d] * tp_w1[d * C_ + tid];
        s_t1[tid] = 0.5f * acc * (1.0f + erff(acc * 0.70710678118654752f));
    }
    // --- gate scalar dots (2 per batch) ---
    if (tid < 2) {
        const float* f = (tid == 1) ? (F_teacher + b * DT_) : (F_unified + b * DT_);
        float acc = fa_b[0];
        for (int d = 0; d < DT_; ++d) acc += f[d] * fa_w[d];
        s_fs[tid] = acc;
    }
    __syncthreads();

    // --- t = t1 @ tp_w2 + b2 ---
    {
        float acc = tp_b2[tid];
        for (int d = 0; d < C_; ++d) acc += s_t1[d] * tp_w2[d * C_ + tid];
        s_t[tid] = acc;
    }
    __syncthreads();

    // --- q = t @ q_w + q_b ---
    {
        float acc = q_b[tid];
        for (int d = 0; d < C_; ++d) acc += s_t[d] * q_w[d * C_ + tid];
        s_q[tid] = acc;
    }
    __syncthreads();

    // --- w_qk[b][h][c] and b_qk[b][h] ---
    for (int h = 0; h < NH_; ++h) {
        float acc = 0.0f;
        const float* qr = &s_q[h * HD_];
        const float* kr = &k_w[tid * C_ + h * HD_];
        #pragma unroll 8
        for (int j = 0; j < HD_; ++j) acc += qr[j] * kr[j];
        ws_wqk[(b * NH_ + h) * C_ + tid] = acc;
    }
    if (tid < NH_) {
        float acc = 0.0f;
        for (int j = 0; j < HD_; ++j) acc += s_q[tid * HD_ + j] * k_b[tid * HD_ + j];
        ws_bqk[b * NH_ + tid] = acc;
    }

    // --- gate g[b][c] ---
    {
        float a0 = 1.0f / (1.0f + __expf(s_fs[1] - s_fs[0]));   // softmax over 2
        const float* fu  = F_unified + b * DT_;
        const float* ftc = F_teacher + b * DT_;
        float gu = 0.0f, gt = 0.0f;
        for (int d = 0; d < DT_; ++d) {
            float w = gp_w[d * C_ + tid];
            gu += fu[d] * w;
            gt += ftc[d] * w;
        }
        float v = a0 * gu + (1.0f - a0) * gt + gp_b[tid];
        ws_g[b * C_ + tid] = 1.0f / (1.0f + __expf(-v));
    }
}

// ---------------------------------------------------------------------------
// K2: scores[b,h,t] = (F_vis[b,:,t]·w_qk[b,h,:] + b_qk[b,h]) * hd^-0.5
// via V_WMMA_F32_16X16X4_F32:  D[16(M=heads) x 16(N=t)] += A[16x4(c)] * B[4(c)x16(t)]
// A layout (f32 16x4): lanes0-15 M=lane,{K=k0,k0+1}; lanes16-31 M=lane-16,{K=k0+2,k0+3}
// B layout (f32 4x16): lanes0-15 N=lane,{K=k0,k0+1}; lanes16-31 N=lane-16,{K=k0+2,k0+3}
// F_vis rows are contiguous in t, so B loads are natural (no transpose).
// A is staged in LDS zero-padded to 16 head rows, so no per-iteration masking
// sits on the ds_load -> v_wmma dependent chain.
// ---------------------------------------------------------------------------
__global__ __launch_bounds__(256) void k2_scores(
    const float* __restrict__ Fv, const float* __restrict__ ws_wqk,
    const float* __restrict__ ws_bqk, float* __restrict__ ws_scores)
{
    __shared__ float s_wqk[16 * C_];    // 16 KB: heads 0..7 real, 8..15 zeros
    __shared__ float s_bqk[NH_];
    const int tid = threadIdx.x;
    const int b   = blockIdx.x >> 5;      // /32
    const int tz  = blockIdx.x & 31;

    for (int idx = tid; idx < 16 * C_; idx += 256)
        s_wqk[idx] = (idx < NH_ * C_) ? ws_wqk[b * NH_ * C_ + idx] : 0.0f;
    if (tid < NH_) s_bqk[tid] = ws_bqk[b * NH_ + tid];
    __syncthreads();

    const int lane = tid & 31;
    const int wave = tid >> 5;
    const int t0   = tz * 128 + wave * 16;           // 16-token tile per wave
    const float* fb = Fv + (size_t)b * C_ * T_;

    const int m    = lane & 15;                      // head row (8..15 -> zeros)
    const int n    = lane & 15;                      // token col within tile
    const int koff = (lane >> 4) << 1;               // 0 or 2
    const float* arow = &s_wqk[m * C_];

    union { v8f v; float f[8]; } acc;
    #pragma unroll
    for (int j = 0; j < 8; ++j) acc.f[j] = 0.0f;

    for (int k0 = 0; k0 < C_; k0 += 4) {
        v2f a, bb;
        a.x = arow[k0 + koff];
        a.y = arow[k0 + koff + 1];
        bb.x = fb[(size_t)(k0 + koff)     * T_ + t0 + n];
        bb.y = fb[(size_t)(k0 + koff + 1) * T_ + t0 + n];
        acc.v = __builtin_amdgcn_wmma_f32_16x16x4_f32(
                    false, a, false, bb, (short)0, acc.v, false, false);
    }

    // D: VGPR j, lanes0-15 -> M=j (head j), N=lane.  lanes16-31 hold M=8+j (unused).
    const float scale = 0.17677669529663689f;        // 32^-0.5
    if (lane < 16) {
        int t = t0 + lane;
        #pragma unroll
        for (int j = 0; j < NH_; ++j)
            ws_scores[((size_t)(b * NH_ + j)) * T_ + t] = (acc.f[j] + s_bqk[j]) * scale;
    }
}

// ---------------------------------------------------------------------------
// K3: softmax over T per (b,h); writes attn directly into d_out's attn region.
// float4 (b128) traffic; rows are 16 KB and 16B-aligned.
// ---------------------------------------------------------------------------
__global__ __launch_bounds__(256) void k3_softmax(
    const float* __restrict__ ws_scores, float* __restrict__ attn)
{
    __shared__ float red[256];
    const int tid = threadIdx.x;
    const float4* row4 = (const float4*)(ws_scores + (size_t)blockIdx.x * T_);

    float m = -1e30f;
    for (int i = tid; i < T_ / 4; i += 256) {
        float4 v = row4[i];
        m = fmaxf(m, fmaxf(fmaxf(v.x, v.y), fmaxf(v.z, v.w)));
    }
    red[tid] = m; __syncthreads();
    for (int s = 128; s > 0; s >>= 1) {
        if (tid < s) red[tid] = fmaxf(red[tid], red[tid + s]);
        __syncthreads();
    }
    const float M = red[0];
    __syncthreads();

    float sum = 0.0f;
    for (int i = tid; i < T_ / 4; i += 256) {
        float4 v = row4[i];
        sum += __expf(v.x - M) + __expf(v.y - M) + __expf(v.z - M) + __expf(v.w - M);
    }
    red[tid] = sum; __syncthreads();
    for (int s = 128; s > 0; s >>= 1) {
        if (tid < s) red[tid] += red[tid + s];
        __syncthreads();
    }
    const float inv = 1.0f / red[0];

    float4* arow4 = (float4*)(attn + (size_t)blockIdx.x * T_);
    for (int i = tid; i < T_ / 4; i += 256) {
        float4 v = row4[i], o;
        o.x = __expf(v.x - M) * inv;
        o.y = __expf(v.y - M) * inv;
        o.z = __expf(v.z - M) * inv;
        o.w = __expf(v.w - M) * inv;
        arow4[i] = o;
    }
}

// ---------------------------------------------------------------------------
// K4: xa[b,h,c] = sum_t attn[b,h,t] * F_vis[b,c,t].  One block per (b,c).
// float4 loads for both F_vis (L2-hot after K2) and attn (L2-resident).
// ---------------------------------------------------------------------------
__global__ __launch_bounds__(256) void k4_xa(
    const float* __restrict__ Fv, const float* __restrict__ attn,
    float* __restrict__ ws_xa)
{
    __shared__ float red[NH_ * 256];     // 8 KB
    const int tid = threadIdx.x;
    const int b = blockIdx.x >> 8;
    const int c = blockIdx.x & 255;
    const float4* fr4 = (const float4*)(Fv + ((size_t)b * C_ + c) * T_);
    const float* ab   = attn + (size_t)b * NH_ * T_;

    float acc[NH_];
    #pragma unroll
    for (int h = 0; h < NH_; ++h) acc[h] = 0.0f;

    for (int i = tid; i < T_ / 4; i += 256) {
        float4 v = fr4[i];
        #pragma unroll
        for (int h = 0; h < NH_; ++h) {
            float4 a = ((const float4*)(ab + (size_t)h * T_))[i];
            acc[h] += v.x * a.x + v.y * a.y + v.z * a.z + v.w * a.w;
        }
    }
    #pragma unroll
    for (int h = 0; h < NH_; ++h) red[h * 256 + tid] = acc[h];
    __syncthreads();
    for (int s = 128; s > 0; s >>= 1) {
        if (tid < s) {
            #pragma unroll
            for (int h = 0; h < NH_; ++h) red[h * 256 + tid] += red[h * 256 + tid + s];
        }
        __syncthreads();
    }
    if (tid < NH_)
        ws_xa[((size_t)b * NH_ + tid) * C_ + c] = red[tid * 256];
}

// ---------------------------------------------------------------------------
// K5: z[b,c] = xa[b,c/32,:]·v_w[:,c] + v_b[c]; zo = z@o_w+o_b; gz = g*zo.
// One block per batch.
// ---------------------------------------------------------------------------
__global__ __launch_bounds__(256) void k5_zproj(
    const float* __restrict__ ws_xa, const float* __restrict__ v_w,
    const float* __restrict__ v_b,   const float* __restrict__ o_w,
    const float* __restrict__ o_b,   const float* __restrict__ ws_g,
    float* __restrict__ ws_gz)
{
    __shared__ float s_z[C_];
    const int b = blockIdx.x;
    const int c = threadIdx.x;

    {
        const int h = c >> 5;
        const float* xr = &ws_xa[((size_t)b * NH_ + h) * C_];
        float acc = v_b[c];
        for (int cc = 0; cc < C_; ++cc) acc += xr[cc] * v_w[cc * C_ + c];
        s_z[c] = acc;
    }
    __syncthreads();
    {
        float acc = o_b[c];
        for (int cc = 0; cc < C_; ++cc) acc += s_z[cc] * o_w[cc * C_ + c];
        ws_gz[b * C_ + c] = ws_g[b * C_ + c] * acc;
    }
}

// ---------------------------------------------------------------------------
// K6: y[b,t,c] = F_vis[b,c,t] + gz[b,c]; LayerNorm over c (biased var);
// out[b,c,t] = (y-mu)*rsqrt(var+eps)*ln_g[c] + ln_b[c].
// Tile 256c x 32t in LDS (32 KB); b128 global traffic both directions.
// ---------------------------------------------------------------------------
__global__ __launch_bounds__(256) void k6_fuse(
    const float* __restrict__ Fv, const float* __restrict__ ws_gz,
    const float* __restrict__ ln_g, const float* __restrict__ ln_b,
    float* __restrict__ out)
{
    __shared__ float tile[C_ * 32];      // 32 KB
    __shared__ float s_gz[C_], s_lg[C_], s_lb[C_];
    __shared__ float ps[256], ps2[256];
    __shared__ float s_mu[32], s_rs[32];

    const int tid = threadIdx.x;
    const int b  = blockIdx.x >> 7;               // T_/32 = 128 tiles per batch
    const int t0 = (blockIdx.x & 127) * 32;

    s_gz[tid] = ws_gz[b * C_ + tid];
    s_lg[tid] = ln_g[tid];
    s_lb[tid] = ln_b[tid];
    __syncthreads();

    const float* fb = Fv + (size_t)b * C_ * T_ + t0;
    for (int idx = tid; idx < C_ * 8; idx += 256) {      // 8 x float4 per channel row
        int c = idx >> 3, q = idx & 7;
        float4 v = *(const float4*)(fb + (size_t)c * T_ + q * 4);
        float g = s_gz[c];
        v.x += g; v.y += g; v.z += g; v.w += g;
        *(float4*)(&tile[c * 32 + q * 4]) = v;
    }
    __syncthreads();

    {   // partial stats: 8 threads per token, 32 channels each
        int tt = tid & 31, seg = tid >> 5;
        float s = 0.0f, s2 = 0.0f;
        #pragma unroll 8
        for (int i = 0; i < 32; ++i) {
            float v = tile[(seg * 32 + i) * 32 + tt];
            s += v; s2 += v * v;
        }
        ps[tid] = s; ps2[tid] = s2;
    }
    __syncthreads();
    if (tid < 32) {
        float S = 0.0f, S2 = 0.0f;
        #pragma unroll
        for (int k = 0; k < 8; ++k) { S += ps[k * 32 + tid]; S2 += ps2[k * 32 + tid]; }
        float mu  = S * (1.0f / 256.0f);
        float var = S2 * (1.0f / 256.0f) - mu * mu;
        s_mu[tid] = mu;
        s_rs[tid] = rsqrtf(var + EPS_);
    }
    __syncthreads();

    float* ob = out + (size_t)b * C_ * T_ + t0;
    for (int idx = tid; idx < C_ * 8; idx += 256) {
        int c = idx >> 3, q = idx & 7;
        float4 v = *(const float4*)(&tile[c * 32 + q * 4]);
        float lg = s_lg[c], lb = s_lb[c];
        int tt = q * 4;
        v.x = (v.x - s_mu[tt + 0]) * s_rs[tt + 0] * lg + lb;
        v.y = (v.y - s_mu[tt + 1]) * s_rs[tt + 1] * lg + lb;
        v.z = (v.z - s_mu[tt + 2]) * s_rs[tt + 2] * lg + lb;
        v.w = (v.w - s_mu[tt + 3]) * s_rs[tt + 3] * lg + lb;
        *(float4*)(ob + (size_t)c * T_ + q * 4) = v;
    }
}

// ---------------------------------------------------------------------------
extern "C" void kernel_launch(void* const* d_in, const int* in_sizes, int n_in,
                              void* d_out, int out_size, void* d_ws, size_t ws_size,
                              hipStream_t stream)
{
    const float* F_vis     = (const float*)d_in[0];
    const float* F_text    = (const float*)d_in[1];
    const float* F_unified = (const float*)d_in[2];
    const float* F_teacher = (const float*)d_in[3];
    const float* tp_w1 = (const float*)d_in[4];
    const float* tp_b1 = (const float*)d_in[5];
    const float* tp_w2 = (const float*)d_in[6];
    const float* tp_b2 = (const float*)d_in[7];
    const float* q_w  = (const float*)d_in[8];
    const float* q_b  = (const float*)d_in[9];
    const float* k_w  = (const float*)d_in[10];
    const float* k_b  = (const float*)d_in[11];
    const float* v_w  = (const float*)d_in[12];
    const float* v_b  = (const float*)d_in[13];
    const float* o_w  = (const float*)d_in[14];
    const float* o_b  = (const float*)d_in[15];
    const float* fa_w = (const float*)d_in[16];
    const float* fa_b = (const float*)d_in[17];
    const float* gp_w = (const float*)d_in[18];
    const float* gp_b = (const float*)d_in[19];
    const float* ln_g = (const float*)d_in[20];
    const float* ln_b = (const float*)d_in[21];

    float* out  = (float*)d_out;
    float* attn = out + (size_t)B_ * C_ * T_;     // second tuple output [B,NH,1,T]

    float* ws      = (float*)d_ws;
    float* ws_wqk  = ws;            // [B][NH][C]   32768
    float* ws_bqk  = ws + 32768;    // [B][NH]        128
    float* ws_g    = ws + 32896;    // [B][C]        4096
    float* ws_gz   = ws + 36992;    // [B][C]        4096
    float* ws_xa   = ws + 41088;    // [B][NH][C]   32768
    float* ws_sc   = ws + 73856;    // [B][NH][T]  524288

    k1_text<<<B_, 256, 0, stream>>>(F_text, F_unified, F_teacher,
                                    tp_w1, tp_b1, tp_w2, tp_b2,
                                    q_w, q_b, k_w, k_b, fa_w, fa_b, gp_w, gp_b,
                                    ws_wqk, ws_bqk, ws_g);
    k2_scores<<<B_ * (T_ / 128), 256, 0, stream>>>(F_vis, ws_wqk, ws_bqk, ws_sc);
    k3_softmax<<<B_ * NH_, 256, 0, stream>>>(ws_sc, attn);
    k4_xa<<<B_ * C_, 256, 0, stream>>>(F_vis, attn, ws_xa);
    k5_zproj<<<B_, 256, 0, stream>>>(ws_xa, v_w, v_b, o_w, o_b, ws_g, ws_gz);
    k6_fuse<<<B_ * (T_ / 32), 256, 0, stream>>>(F_vis, ws_gz, ln_g, ln_b, out);
}